// LocalKernelPointEvalPeriodic_26817775796359
// MI455X (gfx1250) — compile-verified
//
#include <hip/hip_runtime.h>

// LocalKernelPointEvalPeriodic — B=256 batches, 33x33 bilinear-sampled patch
// (circular pad 16) dotted with a 33x33 kernel. Latency-bound (~1.2 MB touched,
// ~50 ns of HBM time at 23.3 TB/s), so: one workgroup per batch, one pass.
// Per-batch bilinear weights are constant (integer tap offsets), and the
// reference's clip never binds, so each output is a weighted sum over a
// contiguous 34x34 wrap-indexed window. Window staged to LDS once; taps read
// from LDS; wave-level reduction done with V_WMMA_F32_16X16X4_F32 (f32-exact).

typedef __attribute__((ext_vector_type(2))) float v2f;
typedef __attribute__((ext_vector_type(8))) float v8f;

__global__ __launch_bounds__(256)
void lkpe_kernel(const float* __restrict__ m,       // [B,1,H,W]
                 const float* __restrict__ coords,  // [B,2] (x,y)
                 const float* __restrict__ kern,    // [K,K]
                 float* __restrict__ out)           // [B,1]
{
    constexpr int H = 1024, W = 1024, K = 33, KH = 16;
    constexpr int WIN = K + 1;  // 34

    __shared__ float sv[WIN][WIN + 1];  // +1 pad vs bank conflicts
    __shared__ float sk[K * K];
    __shared__ float swave[8];

    const int b   = blockIdx.x;
    const int tid = threadIdx.x;
    const float* __restrict__ img = m + (size_t)b * (size_t)(H * W);

    const float cx = coords[2 * b + 0];
    const float cy = coords[2 * b + 1];
    const int ix0 = (int)floorf(cx);
    const int iy0 = (int)floorf(cy);
    const float wx = cx - (float)ix0;
    const float wy = cy - (float)iy0;

    // Stage the 34x34 window; wrap indexing resolved once here.
    for (int p = tid; p < WIN * WIN; p += 256) {
        int j = p / WIN, i = p - j * WIN;
        int yy = iy0 - KH + j;             // in [-16, H+15]
        yy += (yy < 0) ? H : 0;
        yy -= (yy >= H) ? H : 0;
        int xx = ix0 - KH + i;
        xx += (xx < 0) ? W : 0;
        xx -= (xx >= W) ? W : 0;
        sv[j][i] = img[(size_t)yy * W + xx];
    }
    for (int p = tid; p < K * K; p += 256) sk[p] = kern[p];
    __syncthreads();

    const float w00 = (1.f - wx) * (1.f - wy);
    const float w01 = wx * (1.f - wy);
    const float w10 = (1.f - wx) * wy;
    const float w11 = wx * wy;

    // Each thread handles ~4-5 of the 1089 taps.
    float acc = 0.f;
    for (int t = tid; t < K * K; t += 256) {
        int ky = t / K, kx = t - ky * K;
        float v00 = sv[ky][kx];
        float v01 = sv[ky][kx + 1];
        float v10 = sv[ky + 1][kx];
        float v11 = sv[ky + 1][kx + 1];
        float patch = v00 * w00 + v01 * w01 + v10 * w10 + v11 * w11;
        acc = fmaf(patch, sk[t], acc);
    }

    // Intra-wave reduction via f32 WMMA (exact in f32):
    // A[m,k]: lanes 0-15 -> rows m with K=0,1 ; lanes 16-31 -> rows m with K=2,3.
    // a = {acc, 0} => A[m,0]=p[m], A[m,2]=p[m+16], others 0. B = all ones.
    // D[m,n] = p[m] + p[m+16] for every n.
    v2f a;  a.x = acc; a.y = 0.f;
    v2f bo; bo.x = 1.f; bo.y = 1.f;
    v8f c = {};
    c = __builtin_amdgcn_wmma_f32_16x16x4_f32(
            /*neg_a=*/false, a, /*neg_b=*/false, bo,
            /*c_mod=*/(short)0, c, /*reuse_a=*/false, /*reuse_b=*/false);

    // Lane 0 holds D[0..7,0] in c[0..7]; lane 16 holds D[8..15,0].
    float tot = 0.f;
#pragma unroll
    for (int r = 0; r < 8; ++r) tot += c[r];
    float other = __shfl(tot, 16, 32);  // wave32
    if ((tid & 31) == 0) swave[tid >> 5] = tot + other;
    __syncthreads();

    if (tid == 0) {
        float r = 0.f;
#pragma unroll
        for (int w = 0; w < 8; ++w) r += swave[w];
        out[b] = r;
    }
}

extern "C" void kernel_launch(void* const* d_in, const int* in_sizes, int n_in,
                              void* d_out, int out_size, void* d_ws, size_t ws_size,
                              hipStream_t stream) {
    const float* m      = (const float*)d_in[0];  // [B,1,H,W] f32
    const float* coords = (const float*)d_in[1];  // [B,2]     f32
    const float* kern   = (const float*)d_in[2];  // [33,33]   f32
    float* out = (float*)d_out;                   // [B,1]     f32

    const int B = in_sizes[1] / 2;                // 256
    lkpe_kernel<<<B, 256, 0, stream>>>(m, coords, kern, out);
}